// YOLO_V1_LOSS_65472481460515
// MI455X (gfx1250) — compile-verified
//
#include <hip/hip_runtime.h>

// ---------------------------------------------------------------------------
// YOLO v1 loss, MI455X (gfx1250, wave32).
// Memory-bound streaming reduction: 96.4 MB in -> 6 scalars out.
// - each lane processes 2 adjacent cells (240B, 16B-aligned) -> global_load_b128
// - per-lane fp32 accumulation over grid-stride loop
// - cross-lane reduction seeded by V_WMMA_F32_16X16X4_F32 (fp32-exact),
//   finished with shfl_xor; block reduce in LDS; 6 atomicAdds per block.
// ---------------------------------------------------------------------------

typedef float v2f __attribute__((ext_vector_type(2)));
typedef float v8f __attribute__((ext_vector_type(8)));

#define NCLS 20
#define L_COORD 5.0f
#define L_NOOBJ 0.5f
#define IMG 448.0f
#define IMGM1 447.0f
#define GRIDF 64.0f

__device__ __forceinline__ float iou_box(float bx, float by, float bw, float bh,
                                         float gx, float gy,
                                         float g5, float g6, float g7, float g8,
                                         float g9) {
  float px = truncf((bx + gx) * GRIDF);
  float py = truncf((by + gy) * GRIDF);
  float pw = truncf(bw * IMG);
  float ph = truncf(bh * IMG);
  float x1 = fmaxf(0.0f, px - pw * 0.5f);
  float y1 = fmaxf(0.0f, py - ph * 0.5f);
  float x2 = fminf(IMGM1, px + pw * 0.5f);
  float y2 = fminf(IMGM1, py + ph * 0.5f);
  float parea = (x2 - x1) * (y2 - y1);
  float lx = fmaxf(x1, g5);
  float rx = fminf(x2, g7);
  float uy = fmaxf(y1, g6);
  float dy = fminf(y2, g8);
  float inter = (rx - lx) * (dy - uy);
  bool valid = (rx >= lx) && (dy >= uy);
  float io = inter / (parea + g9 - inter);
  return valid ? io : 0.0f;
}

__global__ void yolo_zero6(float* out) {
  if (threadIdx.x < 6) out[threadIdx.x] = 0.0f;
}

__global__ __launch_bounds__(256) void yolo_v1_loss_kernel(
    const float* __restrict__ bb, const float* __restrict__ gt,
    float* __restrict__ out, int npairs) {
  float a0 = 0.f, a1 = 0.f, a2 = 0.f, a3 = 0.f, a4 = 0.f, a5 = 0.f;

  const int tid = blockIdx.x * blockDim.x + threadIdx.x;
  const int stride = gridDim.x * blockDim.x;

  for (int p = tid; p < npairs; p += stride) {
    // 2 cells per lane: 240B per buffer, 16-byte aligned -> b128 loads.
    const float4* b4 = reinterpret_cast<const float4*>(bb) + (size_t)p * 15;
    const float4* g4 = reinterpret_cast<const float4*>(gt) + (size_t)p * 15;
    float bv[60], gv[60];
#pragma unroll
    for (int i = 0; i < 15; ++i) {
      float4 t = b4[i];
      bv[4 * i + 0] = t.x; bv[4 * i + 1] = t.y;
      bv[4 * i + 2] = t.z; bv[4 * i + 3] = t.w;
    }
#pragma unroll
    for (int i = 0; i < 15; ++i) {
      float4 t = g4[i];
      gv[4 * i + 0] = t.x; gv[4 * i + 1] = t.y;
      gv[4 * i + 2] = t.z; gv[4 * i + 3] = t.w;
    }

    const int c0 = 2 * p;
#pragma unroll
    for (int h = 0; h < 2; ++h) {
      const int c = c0 + h;
      const int bidx = c / 49;
      const int rem = c - 49 * bidx;
      const int gyi = rem / 7;
      const int gxi = rem - 7 * gyi;
      const float gx = (float)gxi, gy = (float)gyi;
      const float* B = bv + 30 * h;
      const float* G = gv + 30 * h;

      float iou1 = iou_box(B[0], B[1], B[2], B[3], gx, gy,
                           G[5], G[6], G[7], G[8], G[9]);
      float iou2 = iou_box(B[5], B[6], B[7], B[8], gx, gy,
                           G[5], G[6], G[7], G[8], G[9]);
      bool r1 = iou1 > iou2;

      float pX = r1 ? B[0] : B[5];
      float pY = r1 ? B[1] : B[6];
      float pW = r1 ? B[2] : B[7];
      float pH = r1 ? B[3] : B[8];
      float pC = r1 ? B[4] : B[9];
      float nC = r1 ? B[9] : B[4];
      float iou = r1 ? iou1 : iou2;

      float fm = (rintf(G[4]) != 0.0f) ? 1.0f : 0.0f;

      float noobj = L_NOOBJ * (B[4] * B[4] + B[9] * B[9]);

      float dx = pX - G[0];
      float dy = pY - G[1];
      float dw = sqrtf(pW) - sqrtf(G[2]);
      float dh = sqrtf(pH) - sqrtf(G[3]);
      float coord = L_COORD * (dx * dx + dy * dy + dw * dw + dh * dh);

      float dc = pC - iou;
      float conf = dc * dc + L_NOOBJ * nC * nC;

      float cls = 0.0f;
#pragma unroll
      for (int i = 0; i < NCLS; ++i) {
        float d = G[10 + i] - B[10 + i];
        cls = fmaf(d, d, cls);
      }

      float om = 1.0f - fm;
      a0 += fm * (coord + conf + cls) + om * noobj;
      a1 += fm * coord;
      a2 += fm * conf + om * noobj;
      a3 += fm * cls;
      a4 += fm * iou;
      a5 += fm;
    }
  }

  // -------- cross-lane reduction, seeded by fp32 WMMA (16x16x4) ----------
  // A layout (16x4 f32): lanes 0-15 hold {K0,K1}, lanes 16-31 hold {K2,K3},
  // row M = lane%16. Selector rows: A[0][*]=[1,0,1,0], A[1][*]=[0,1,0,1],
  // rows >=2 zero. B layout (4x16): lane n holds {B[0][n],B[1][n]} (n<16) and
  // {B[2][n-16],B[3][n-16]} (n>=16). Result: D[0][n] = accX(n)+accX(n+16),
  // D[1][n] = accY(n)+accY(n+16) -> one WMMA folds 32->16 lanes for 2 accs.
  const int lane = threadIdx.x & 31;
  const int wave = threadIdx.x >> 5;

  v2f A;
  A[0] = ((lane & 15) == 0) ? 1.0f : 0.0f;
  A[1] = ((lane & 15) == 1) ? 1.0f : 0.0f;
  v8f Z = {0.f, 0.f, 0.f, 0.f, 0.f, 0.f, 0.f, 0.f};

  v2f b01; b01[0] = a0; b01[1] = a1;
  v2f b23; b23[0] = a2; b23[1] = a3;
  v2f b45; b45[0] = a4; b45[1] = a5;

  v8f d01 = __builtin_amdgcn_wmma_f32_16x16x4_f32(false, A, false, b01,
                                                  (short)0, Z, false, false);
  v8f d23 = __builtin_amdgcn_wmma_f32_16x16x4_f32(false, A, false, b23,
                                                  (short)0, Z, false, false);
  v8f d45 = __builtin_amdgcn_wmma_f32_16x16x4_f32(false, A, false, b45,
                                                  (short)0, Z, false, false);

  float p0 = d01[0], p1 = d01[1];
  float p2 = d23[0], p3 = d23[1];
  float p4 = d45[0], p5 = d45[1];

  // lanes 16-31 carry zeros (rows M=8,9 of D); xor steps stay inside halves,
  // lane 0 ends with the full 32-lane sum of each accumulator.
#pragma unroll
  for (int m = 1; m < 16; m <<= 1) {
    p0 += __shfl_xor(p0, m, 32);
    p1 += __shfl_xor(p1, m, 32);
    p2 += __shfl_xor(p2, m, 32);
    p3 += __shfl_xor(p3, m, 32);
    p4 += __shfl_xor(p4, m, 32);
    p5 += __shfl_xor(p5, m, 32);
  }

  __shared__ float red[8 * 6];
  if (lane == 0) {
    red[wave * 6 + 0] = p0;
    red[wave * 6 + 1] = p1;
    red[wave * 6 + 2] = p2;
    red[wave * 6 + 3] = p3;
    red[wave * 6 + 4] = p4;
    red[wave * 6 + 5] = p5;
  }
  __syncthreads();

  if (threadIdx.x < 6) {
    float s = 0.0f;
    const int nw = blockDim.x >> 5;
    for (int w = 0; w < nw; ++w) s += red[w * 6 + threadIdx.x];
    atomicAdd(&out[threadIdx.x], s);
  }
}

extern "C" void kernel_launch(void* const* d_in, const int* in_sizes, int n_in,
                              void* d_out, int out_size, void* d_ws,
                              size_t ws_size, hipStream_t stream) {
  const float* bb = (const float*)d_in[0];   // (8192,7,7,30) f32
  const float* gt = (const float*)d_in[1];   // (8192,7,7,30) f32
  float* out = (float*)d_out;                // 6 scalars

  const int ncells = in_sizes[0] / 30;       // 401408
  const int npairs = ncells / 2;             // 200704 (even by construction)

  yolo_zero6<<<1, 32, 0, stream>>>(out);

  const int threads = 256;
  const int blocks = (npairs + threads - 1) / threads;  // 784
  yolo_v1_loss_kernel<<<blocks, threads, 0, stream>>>(bb, gt, out, npairs);
}